// HyperbolicEncoder_50792283242942
// MI455X (gfx1250) — compile-verified
//
#include <hip/hip_runtime.h>
#include <hip/hip_bf16.h>

typedef __attribute__((ext_vector_type(2))) float v2f;
typedef __attribute__((ext_vector_type(8))) float v8f;

#define N_NODES 50000
#define N_EDGES 800000
#define NODE_DIM 256
#define HID 128
#define EMB 64

// ---------- helpers ----------
__device__ __forceinline__ float atanh_c(float x) {           // x in [0, 0.99]
    return 0.5f * __logf((1.f + x) / (1.f - x));
}
__device__ __forceinline__ unsigned f2ord(float f) {          // order-preserving encode
    unsigned u = __float_as_uint(f);
    return (u & 0x80000000u) ? ~u : (u | 0x80000000u);
}
__device__ __forceinline__ float ord2f(unsigned u) {
    return (u & 0x80000000u) ? __uint_as_float(u & 0x7FFFFFFFu) : __uint_as_float(~u);
}
__device__ __forceinline__ float wave_sum(float v) {
    #pragma unroll
    for (int off = 16; off; off >>= 1) v += __shfl_xor(v, off, 32);
    return v;
}

// ---------- fp32 WMMA GEMM: H = act(X @ W + b), X[M,K] row-major, W[K,N] row-major ----------
// grid.x = M/16 ; block = 32 * (N/16) threads ; wave w owns 16x16 output tile at col 16*w.
template<int KDIM, int NDIM, bool RELU>
__global__ void gemm_wmma_f32(const float* __restrict__ X, const float* __restrict__ W,
                              const float* __restrict__ bias, float* __restrict__ H) {
    const int mBase = blockIdx.x * 16;
    const int wave  = threadIdx.x >> 5;
    const int lane  = threadIdx.x & 31;
    const int nBase = wave * 16;
    const int r     = lane & 15;   // M index (A) / N index (B,C,D)
    const int hi    = lane >> 4;   // K-half selector for A/B, M-half for C/D

    v8f c = {};
    const float* __restrict__ Arow = X + (size_t)(mBase + r) * KDIM;

    #pragma unroll 4
    for (int k0 = 0; k0 < KDIM; k0 += 4) {
        const int kk = k0 + 2 * hi;
        v2f a, b;
        a.x = Arow[kk];
        a.y = Arow[kk + 1];
        b.x = W[(size_t)kk * NDIM + nBase + r];
        b.y = W[(size_t)(kk + 1) * NDIM + nBase + r];
        c = __builtin_amdgcn_wmma_f32_16x16x4_f32(false, a, false, b, (short)0, c, false, false);
    }

    const float bc = bias[nBase + r];
    float* __restrict__ Hp = H + nBase + r;
    #pragma unroll
    for (int j = 0; j < 8; ++j) {
        const int m = mBase + j + 8 * hi;
        float v = c[j] + bc;
        if (RELU) v = fmaxf(v, 0.f);
        Hp[(size_t)m * NDIM] = v;
    }
}

// ---------- row-wise maps (one wave32 per row) ----------
template<int D>
__global__ void row_expmap(const float* __restrict__ in, float* __restrict__ out, int n) {
    const int node = blockIdx.x * (blockDim.x >> 5) + (threadIdx.x >> 5);
    const int lane = threadIdx.x & 31;
    if (node >= n) return;
    constexpr int PER = D / 32;
    float v[PER]; float ss = 0.f;
    #pragma unroll
    for (int i = 0; i < PER; ++i) { v[i] = in[(size_t)node * D + lane * PER + i]; ss += v[i] * v[i]; }
    const float norm = sqrtf(wave_sum(ss));
    const float f = (norm > 0.f) ? (tanhf(norm) / norm) : 1.f;
    #pragma unroll
    for (int i = 0; i < PER; ++i) out[(size_t)node * D + lane * PER + i] = v[i] * f;
}

template<int D>
__global__ void row_logmap(const float* __restrict__ in, float* __restrict__ out, int n) {
    const int node = blockIdx.x * (blockDim.x >> 5) + (threadIdx.x >> 5);
    const int lane = threadIdx.x & 31;
    if (node >= n) return;
    constexpr int PER = D / 32;
    float v[PER]; float ss = 0.f;
    #pragma unroll
    for (int i = 0; i < PER; ++i) { v[i] = in[(size_t)node * D + lane * PER + i]; ss += v[i] * v[i]; }
    const float norm = sqrtf(wave_sum(ss));
    const float f = (norm > 0.f) ? (atanh_c(fminf(norm, 0.99f)) / norm) : 1.f;
    #pragma unroll
    for (int i = 0; i < PER; ++i) out[(size_t)node * D + lane * PER + i] = v[i] * f;
}

// logmap0 -> relu -> expmap0 -> logmap0 -> expmap0, fused to scalar scales on relu(logmap0(v))
template<int D>
__global__ void row_relu_chain(const float* __restrict__ in, float* __restrict__ out, int n) {
    const int node = blockIdx.x * (blockDim.x >> 5) + (threadIdx.x >> 5);
    const int lane = threadIdx.x & 31;
    if (node >= n) return;
    constexpr int PER = D / 32;
    float v[PER]; float ss = 0.f;
    #pragma unroll
    for (int i = 0; i < PER; ++i) { v[i] = in[(size_t)node * D + lane * PER + i]; ss += v[i] * v[i]; }
    const float n1 = sqrtf(wave_sum(ss));
    const float l1 = (n1 > 0.f) ? (atanh_c(fminf(n1, 0.99f)) / n1) : 1.f;
    float u[PER]; float ss2 = 0.f;
    #pragma unroll
    for (int i = 0; i < PER; ++i) { u[i] = fmaxf(v[i] * l1, 0.f); ss2 += u[i] * u[i]; }
    const float s = sqrtf(wave_sum(ss2));
    float f = 1.f;
    if (s > 0.f) {
        const float t  = tanhf(s);                     // norm after expmap0
        const float n3 = atanh_c(fminf(t, 0.99f));     // norm after clamped logmap0
        f = tanhf(n3) / s;                             // final expmap0 folded in
    }
    #pragma unroll
    for (int i = 0; i < PER; ++i) out[(size_t)node * D + lane * PER + i] = u[i] * f;
}

// ---------- per-node attention projections: as = h.a[:D], ad = h.a[D:] ----------
template<int D>
__global__ void alpha_kernel(const float* __restrict__ H, const float* __restrict__ aw,
                             float* __restrict__ as, float* __restrict__ ad, int n) {
    const int node = blockIdx.x * (blockDim.x >> 5) + (threadIdx.x >> 5);
    const int lane = threadIdx.x & 31;
    if (node >= n) return;
    float s = 0.f, t = 0.f;
    #pragma unroll
    for (int i = lane; i < D; i += 32) {
        const float h = H[(size_t)node * D + i];
        s += h * aw[i];
        t += h * aw[D + i];
    }
    s = wave_sum(s); t = wave_sum(t);
    if (lane == 0) { as[node] = s; ad[node] = t; }
}

// ---------- edge phase ----------
__global__ void edge_e_kernel(const int* __restrict__ src, const int* __restrict__ dst,
                              const float* __restrict__ as, const float* __restrict__ ad,
                              const float* __restrict__ ab_p, float* __restrict__ e,
                              unsigned* __restrict__ mord, int E) {
    const int k = blockIdx.x * blockDim.x + threadIdx.x;
    if (k >= E) return;
    float v = as[src[k]] + ad[dst[k]] + ab_p[0];
    v = (v > 0.f) ? v : 0.2f * v;                      // leaky_relu(0.2)
    e[k] = v;
    atomicMax(&mord[src[k]], f2ord(v));
}

__global__ void edge_exp_kernel(const int* __restrict__ src, float* __restrict__ e,
                                const unsigned* __restrict__ mord, float* __restrict__ den, int E) {
    const int k = blockIdx.x * blockDim.x + threadIdx.x;
    if (k >= E) return;
    const int s = src[k];
    const float ex = __expf(e[k] - ord2f(mord[s]));
    e[k] = ex;
    atomicAdd(&den[s], ex);
}

// one wave32 per edge: out[dst] += h[src] * att
template<int D>
__global__ void edge_scatter_kernel(const int* __restrict__ src, const int* __restrict__ dst,
                                    const float* __restrict__ ex, const float* __restrict__ den,
                                    const float* __restrict__ H, float* __restrict__ out, int E) {
    const int edge = blockIdx.x * (blockDim.x >> 5) + (threadIdx.x >> 5);
    const int lane = threadIdx.x & 31;
    if (edge >= E) return;
    const int s = src[edge], d = dst[edge];
    const float att = ex[edge] / den[s];
    #pragma unroll
    for (int i = 0; i < D / 32; ++i) {
        const int col = i * 32 + lane;
        atomicAdd(&out[(size_t)d * D + col], H[(size_t)s * D + col] * att);
    }
}

// ---------- launcher ----------
extern "C" void kernel_launch(void* const* d_in, const int* in_sizes, int n_in,
                              void* d_out, int out_size, void* d_ws, size_t ws_size,
                              hipStream_t stream) {
    const float* x    = (const float*)d_in[0];
    const int*   ei   = (const int*)  d_in[1];
    const float* W_in = (const float*)d_in[2];
    const float* b_in = (const float*)d_in[3];
    const float* W1   = (const float*)d_in[4];
    const float* b1   = (const float*)d_in[5];
    const float* a1   = (const float*)d_in[6];
    const float* ba1  = (const float*)d_in[7];
    const float* W2   = (const float*)d_in[8];
    const float* b2   = (const float*)d_in[9];
    const float* a2   = (const float*)d_in[10];
    const float* ba2  = (const float*)d_in[11];

    const int N = N_NODES;
    const int E = in_sizes[1] / 2;
    const int* srcI = ei;
    const int* dstI = ei + E;

    // workspace layout
    float*    bufA = (float*)d_ws;                    // N*HID
    float*    bufB = bufA + (size_t)N * HID;          // N*HID
    float*    bufC = bufB + (size_t)N * HID;          // N*HID
    float*    eBuf = bufC + (size_t)N * HID;          // E
    unsigned* mord = (unsigned*)(eBuf + E);           // N
    float*    den  = (float*)(mord + N);              // N
    float*    as   = den + N;                         // N
    float*    ad   = as + N;                          // N

    const int MT       = N / 16;      // 3125 (exact)
    const int rowBlks  = (N + 7) / 8; // 8 waves/block, 1 wave/row
    const int edgBlks  = (E + 255) / 256;
    const int scatBlks = (E + 7) / 8; // 1 wave/edge

    // Stage 0: h0 = relu(x @ W_in + b_in)            -> A
    gemm_wmma_f32<NODE_DIM, HID, true><<<MT, 256, 0, stream>>>(x, W_in, b_in, bufA);
    // Stage 1: xh = expmap0(h0)                      -> B
    row_expmap<HID><<<rowBlks, 256, 0, stream>>>(bufA, bufB, N);

    // ===== GAT layer 1 (HID -> HID) =====
    row_logmap<HID><<<rowBlks, 256, 0, stream>>>(bufB, bufA, N);                       // xt -> A
    hipMemsetAsync(bufB, 0, (size_t)N * HID * sizeof(float), stream);                  // out accum
    hipMemsetAsync(mord, 0, (size_t)N * sizeof(unsigned), stream);
    hipMemsetAsync(den,  0, (size_t)N * sizeof(float), stream);
    gemm_wmma_f32<HID, HID, false><<<MT, 256, 0, stream>>>(bufA, W1, b1, bufC);        // h1 -> C
    alpha_kernel<HID><<<rowBlks, 256, 0, stream>>>(bufC, a1, as, ad, N);
    edge_e_kernel<<<edgBlks, 256, 0, stream>>>(srcI, dstI, as, ad, ba1, eBuf, mord, E);
    edge_exp_kernel<<<edgBlks, 256, 0, stream>>>(srcI, eBuf, mord, den, E);
    edge_scatter_kernel<HID><<<scatBlks, 256, 0, stream>>>(srcI, dstI, eBuf, den, bufC, bufB, E);
    row_expmap<HID><<<rowBlks, 256, 0, stream>>>(bufB, bufA, N);                       // xh1 -> A

    // Stage 3: fused ReLU wrapper + Dropout wrapper  -> B
    row_relu_chain<HID><<<rowBlks, 256, 0, stream>>>(bufA, bufB, N);

    // ===== GAT layer 2 (HID -> EMB) =====
    row_logmap<HID><<<rowBlks, 256, 0, stream>>>(bufB, bufA, N);                       // xt2 -> A
    hipMemsetAsync(bufB, 0, (size_t)N * EMB * sizeof(float), stream);                  // out accum
    hipMemsetAsync(mord, 0, (size_t)N * sizeof(unsigned), stream);
    hipMemsetAsync(den,  0, (size_t)N * sizeof(float), stream);
    gemm_wmma_f32<HID, EMB, false><<<MT, 128, 0, stream>>>(bufA, W2, b2, bufC);        // h2 -> C
    alpha_kernel<EMB><<<rowBlks, 256, 0, stream>>>(bufC, a2, as, ad, N);
    edge_e_kernel<<<edgBlks, 256, 0, stream>>>(srcI, dstI, as, ad, ba2, eBuf, mord, E);
    edge_exp_kernel<<<edgBlks, 256, 0, stream>>>(srcI, eBuf, mord, den, E);
    edge_scatter_kernel<EMB><<<scatBlks, 256, 0, stream>>>(srcI, dstI, eBuf, den, bufC, bufB, E);
    row_expmap<EMB><<<rowBlks, 256, 0, stream>>>(bufB, (float*)d_out, N);              // final
}